// DistMax_82351702933567
// MI455X (gfx1250) — compile-verified
//
#include <hip/hip_runtime.h>
#include <cstdint>

#define NN 1024
#define DD 256
#define PAD 260   // 260 dwords = 65*16B: keeps float4 alignment, breaks 64-bank conflicts

typedef float v2f __attribute__((ext_vector_type(2)));
typedef float v8f __attribute__((ext_vector_type(8)));

// ---------------- kernel 1: s[i] = dot(xs[i,:], w) ----------------
__global__ __launch_bounds__(256) void sdot_kernel(const float* __restrict__ xs,
                                                   const float* __restrict__ w,
                                                   float* __restrict__ s) {
  const int lane = threadIdx.x & 31;
  const int row  = blockIdx.x * 8 + (threadIdx.x >> 5);
  const float* xr = xs + (size_t)row * DD;
  float p = 0.0f;
  #pragma unroll
  for (int k = 0; k < DD / 32; ++k)
    p = fmaf(xr[lane + 32 * k], w[lane + 32 * k], p);
  #pragma unroll
  for (int off = 16; off > 0; off >>= 1)
    p += __shfl_down(p, off);
  if (lane == 0) s[row] = p;
}

// ---------------- kernel 2: count = sum_g c_g^2 - N ----------------
__global__ __launch_bounds__(1024) void count_kernel(const long long* __restrict__ batch,
                                                     float* __restrict__ cnt_out) {
  __shared__ unsigned h[32];
  if (threadIdx.x < 32) h[threadIdx.x] = 0u;
  __syncthreads();
  atomicAdd(&h[(int)batch[threadIdx.x]], 1u);
  __syncthreads();
  if (threadIdx.x == 0) {
    unsigned long long sum = 0;
    for (int g = 0; g < 32; ++g)
      sum += (unsigned long long)h[g] * (unsigned long long)h[g];
    cnt_out[0] = (float)(sum - (unsigned long long)NN);
  }
}

// ---------------- kernel 3: 16x16 tiles, one wave32 per tile ----------------
__global__ __launch_bounds__(32) void tile_kernel(const float* __restrict__ xs,
                                                  const long long* __restrict__ batch,
                                                  const float* __restrict__ w,
                                                  const float* __restrict__ bptr,
                                                  const float* __restrict__ s,
                                                  float* __restrict__ out0,
                                                  float* __restrict__ maskout) {
  __shared__ __align__(16) float xi[16 * PAD];
  __shared__ __align__(16) float xj[16 * PAD];
  __shared__ __align__(16) float wl[DD];
  __shared__ long long bi[16];
  __shared__ long long bj[16];

  const int lane = threadIdx.x;          // 0..31
  const int j0 = blockIdx.x * 16;
  const int i0 = blockIdx.y * 16;
  const int n  = lane & 15;              // B/C/D column (and A row) for this lane
  const int hi = (lane >> 4) ? 8 : 0;    // C/D: lanes 16-31 hold rows M+8

  if (lane < 16) bi[lane] = batch[i0 + lane];
  else           bj[lane - 16] = batch[j0 + lane - 16];
  __syncthreads();

  // batch is sorted: tile is all-zero iff batch ranges of rows/cols don't overlap
  const bool active = !(bi[15] < bj[0] || bj[15] < bi[0]);

  if (!active) {
    const float4 z = make_float4(0.f, 0.f, 0.f, 0.f);
    #pragma unroll
    for (int k = 0; k < 2; ++k) {
      const int idx = k * 32 + lane;     // 0..63 float4 chunks of the tile
      const int row = idx >> 2;          // 0..15
      const int c4  = idx & 3;
      const size_t off = (size_t)(i0 + row) * NN + (size_t)(j0 + c4 * 4);
      *(float4*)(out0 + off)    = z;
      *(float4*)(maskout + off) = z;
    }
    return;
  }

  // ---- stage 32 rows of xs and 0.5*w into LDS (coalesced float4) ----
  for (int t = lane; t < DD / 4; t += 32) {
    const float4 wv4 = ((const float4*)w)[t];
    wl[t * 4 + 0] = 0.5f * wv4.x;
    wl[t * 4 + 1] = 0.5f * wv4.y;
    wl[t * 4 + 2] = 0.5f * wv4.z;
    wl[t * 4 + 3] = 0.5f * wv4.w;
  }
  for (int t = lane; t < 32 * (DD / 4); t += 32) {
    const int row = t >> 6;              // 0..31
    const int c   = t & 63;              // float4 chunk in row
    const int grow = (row < 16) ? (i0 + row) : (j0 + row - 16);
    const float4 v = ((const float4*)(xs + (size_t)grow * DD))[c];
    float* dst = ((row < 16) ? (xi + row * PAD) : (xj + (row - 16) * PAD)) + c * 4;
    dst[0] = v.x; dst[1] = v.y; dst[2] = v.z; dst[3] = v.w;
  }
  __syncthreads();

  // ---- accumulate C[i,j] = sum_d 0.5*w_d*|x_i[d]-x_j[d]| in WMMA C/D layout ----
  v8f acc = {};
  const float* xjrow = xj + n * PAD;
  for (int d = 0; d < DD; d += 4) {
    const float4 xv = *(const float4*)(xjrow + d);
    const float4 wv = *(const float4*)(wl + d);
    #pragma unroll
    for (int r = 0; r < 8; ++r) {
      const float* xirow = xi + (r + hi) * PAD + d;   // broadcast within half-wave
      acc[r] = fmaf(wv.x, fabsf(xirow[0] - xv.x), acc[r]);
      acc[r] = fmaf(wv.y, fabsf(xirow[1] - xv.y), acc[r]);
      acc[r] = fmaf(wv.z, fabsf(xirow[2] - xv.z), acc[r]);
      acc[r] = fmaf(wv.w, fabsf(xirow[3] - xv.w), acc[r]);
    }
  }

  // ---- rank-2 term via two chained WMMAs, A/B constant across K ----
  // D1 = sum_k (0.125*s_I[m]) * 1           + acc  = 0.5*s_i + acc
  // D2 = sum_k  0.25          * (0.5*s_J[n]+b) + D1 = 0.5*s_j + b + D1
  const float sI = s[i0 + n];              // A row m == lane&15 (both lane halves)
  const float sJ = s[j0 + n];              // B col n == lane&15
  const float bias = bptr[0];

  v2f a1; a1.x = 0.125f * sI; a1.y = 0.125f * sI;
  v2f b1; b1.x = 1.0f;        b1.y = 1.0f;
  v8f d1 = __builtin_amdgcn_wmma_f32_16x16x4_f32(false, a1, false, b1,
                                                 (short)0, acc, false, false);
  v2f a2; a2.x = 0.25f; a2.y = 0.25f;
  const float bv = 0.5f * sJ + bias;
  v2f b2; b2.x = bv; b2.y = bv;
  v8f d2 = __builtin_amdgcn_wmma_f32_16x16x4_f32(false, a2, false, b2,
                                                 (short)0, d1, false, false);

  // ---- mask, relu, store ----
  const long long bjv = bj[n];
  #pragma unroll
  for (int r = 0; r < 8; ++r) {
    const int il = r + hi;
    const int gi = i0 + il;
    const int gj = j0 + n;
    const bool m = (bi[il] == bjv) && (gi != gj);
    const size_t off = (size_t)gi * NN + (size_t)gj;
    out0[off]    = m ? fmaxf(d2[r], 0.0f) : 0.0f;
    maskout[off] = m ? 1.0f : 0.0f;
  }
}

extern "C" void kernel_launch(void* const* d_in, const int* in_sizes, int n_in,
                              void* d_out, int out_size, void* d_ws, size_t ws_size,
                              hipStream_t stream) {
  const float*     xs    = (const float*)d_in[0];
  const long long* batch = (const long long*)d_in[1];
  const float*     w     = (const float*)d_in[2];
  const float*     b     = (const float*)d_in[3];

  float* out0    = (float*)d_out;
  float* maskout = out0 + (size_t)NN * NN;
  float* cnt     = out0 + 2 * (size_t)NN * NN;
  float* s       = (float*)d_ws;            // 1024 floats of scratch

  sdot_kernel<<<NN / 8, 256, 0, stream>>>(xs, w, s);
  count_kernel<<<1, NN, 0, stream>>>(batch, cnt);
  dim3 grid(NN / 16, NN / 16);
  tile_kernel<<<grid, 32, 0, stream>>>(xs, batch, w, b, s, out0, maskout);
}